// EGraphCONV_48077863911783
// MI455X (gfx1250) — compile-verified
//
#include <hip/hip_runtime.h>
#include <hip/hip_bf16.h>

typedef __attribute__((ext_vector_type(2))) float v2f;
typedef __attribute__((ext_vector_type(8))) float v8f;

#define IN_SIZE 128
#define E_SIZE  32
#define H_SIZE  128
#define K_TOT   (IN_SIZE + E_SIZE)   // 160
#define THRESH  0.5f
#define LDS_STRIDE 162               // even (8B-aligned b64 loads), 16 distinct banks over rows

// ---------------- workspace layout (floats) ----------------
// deg      : [0,      N)
// sim_cnt  : [N,     2N)
// esum/agg : [2N,   34N)   (agg overwrites esum after pass 2)
// hk       : [34N,  66N)
// sim_sum  : [66N,  98N)
// hk_norm  : [98N,  99N)

__global__ void k_zero(float* ws, long total) {
    long i = (long)blockIdx.x * blockDim.x + threadIdx.x;
    if (i < total) ws[i] = 0.0f;
}

// pass 1: deg[dst] += 1 ; esum[dst] += edge_feat[e]
__global__ void k_scatter1(const float* __restrict__ ef, const int* __restrict__ dst,
                           float* __restrict__ deg, float* __restrict__ esum, int E) {
    int e = blockIdx.x * blockDim.x + threadIdx.x;
    if (e >= E) return;
    int d = dst[e];
    atomicAdd(&deg[d], 1.0f);
    const float4* row = (const float4*)(ef + (size_t)e * E_SIZE);
    float* es = esum + (size_t)d * E_SIZE;
#pragma unroll
    for (int q = 0; q < 8; ++q) {
        float4 v = row[q];
        atomicAdd(&es[4 * q + 0], v.x);
        atomicAdd(&es[4 * q + 1], v.y);
        atomicAdd(&es[4 * q + 2], v.z);
        atomicAdd(&es[4 * q + 3], v.w);
    }
}

// per node: hk = deg>0 ? esum/deg : 0 ; hk_norm = ||hk||
__global__ void k_hk(const float* __restrict__ deg, const float* __restrict__ esum,
                     float* __restrict__ hk, float* __restrict__ hk_norm, int N) {
    int n = blockIdx.x * blockDim.x + threadIdx.x;
    if (n >= N) return;
    float dg = deg[n];
    float inv = dg > 0.0f ? 1.0f / dg : 0.0f;
    const float* es = esum + (size_t)n * E_SIZE;
    float* h = hk + (size_t)n * E_SIZE;
    float nrm2 = 0.0f;
#pragma unroll
    for (int j = 0; j < E_SIZE; ++j) {
        float v = es[j] * inv;
        h[j] = v;
        nrm2 += v * v;
    }
    hk_norm[n] = sqrtf(nrm2);
}

// pass 2: cos(hk[dst], ef) < 0.5 -> sim_sum[dst]+=ef, sim_cnt[dst]+=1
// mask == (den>0 && num < THRESH*den); den==0 => cos=NaN => mask false (matches ref)
__global__ void k_scatter2(const float* __restrict__ ef, const int* __restrict__ dst,
                           const float* __restrict__ hk, const float* __restrict__ hk_norm,
                           float* __restrict__ sim_sum, float* __restrict__ sim_cnt, int E) {
    int e = blockIdx.x * blockDim.x + threadIdx.x;
    if (e >= E) return;
    int d = dst[e];
    const float4* row = (const float4*)(ef + (size_t)e * E_SIZE);
    const float4* hrow = (const float4*)(hk + (size_t)d * E_SIZE);
    float4 fv[8];
    float num = 0.0f, ef2 = 0.0f;
#pragma unroll
    for (int q = 0; q < 8; ++q) {
        float4 v = row[q];
        float4 h = hrow[q];
        fv[q] = v;
        num += v.x * h.x + v.y * h.y + v.z * h.z + v.w * h.w;
        ef2 += v.x * v.x + v.y * v.y + v.z * v.z + v.w * v.w;
    }
    float den = hk_norm[d] * sqrtf(ef2);
    if (den > 0.0f && num < THRESH * den) {
        atomicAdd(&sim_cnt[d], 1.0f);
        float* ss = sim_sum + (size_t)d * E_SIZE;
#pragma unroll
        for (int q = 0; q < 8; ++q) {
            atomicAdd(&ss[4 * q + 0], fv[q].x);
            atomicAdd(&ss[4 * q + 1], fv[q].y);
            atomicAdd(&ss[4 * q + 2], fv[q].z);
            atomicAdd(&ss[4 * q + 3], fv[q].w);
        }
    }
}

// agg = cnt>0 ? sim_sum/cnt : hk   (written over esum region)
__global__ void k_agg(const float* __restrict__ sim_sum, const float* __restrict__ sim_cnt,
                      const float* __restrict__ hk, float* __restrict__ agg, int N) {
    int n = blockIdx.x * blockDim.x + threadIdx.x;
    if (n >= N) return;
    float c = sim_cnt[n];
    float inv = c > 0.0f ? 1.0f / c : 0.0f;
    const float* ss = sim_sum + (size_t)n * E_SIZE;
    const float* h = hk + (size_t)n * E_SIZE;
    float* a = agg + (size_t)n * E_SIZE;
#pragma unroll
    for (int j = 0; j < E_SIZE; ++j)
        a[j] = c > 0.0f ? ss[j] * inv : h[j];
}

// out[N x 128] = [h_in | agg] @ W^T, f32 WMMA 16x16x4.
// Block: 256 threads = 8 waves; block handles 16 rows; wave w -> cols [16w,16w+16).
__global__ void k_gemm_wmma(const float* __restrict__ h_in, const float* __restrict__ agg,
                            const float* __restrict__ W, float* __restrict__ out, int N) {
    __shared__ float lds[16 * LDS_STRIDE];
    int row0 = blockIdx.x * 16;

    // stage concat tile [16 x 160] into LDS
    for (int idx = threadIdx.x; idx < 16 * K_TOT; idx += 256) {
        int r = idx / K_TOT;
        int c = idx - r * K_TOT;
        int gr = row0 + r;
        float v = (c < IN_SIZE) ? h_in[(size_t)gr * IN_SIZE + c]
                                : agg[(size_t)gr * E_SIZE + (c - IN_SIZE)];
        lds[r * LDS_STRIDE + c] = v;
    }
    __syncthreads();

    int lane = threadIdx.x & 31;       // wave32
    int wave = threadIdx.x >> 5;
    int ln   = lane & 15;              // A: row m ; B: col n
    int half = lane >> 4;              // selects K pair within 16x16x4 fragment
    int col0 = wave * 16;

    const float* wrow = W + (size_t)(col0 + ln) * K_TOT;  // B[k][n] = W[n][k]
    v8f acc = {};
#pragma unroll
    for (int k0 = 0; k0 < K_TOT; k0 += 4) {
        int kk = k0 + 2 * half;
        v2f a = *(const v2f*)&lds[ln * LDS_STRIDE + kk];  // A[m][kk], A[m][kk+1]
        v2f b = *(const v2f*)&wrow[kk];                   // B[kk][n], B[kk+1][n]
        acc = __builtin_amdgcn_wmma_f32_16x16x4_f32(false, a, false, b,
                                                    (short)0, acc, false, false);
    }

    // C/D layout: VGPR r -> M = r + 8*half, N = ln
#pragma unroll
    for (int r = 0; r < 8; ++r)
        out[(size_t)(row0 + r + 8 * half) * H_SIZE + col0 + ln] = acc[r];
}

extern "C" void kernel_launch(void* const* d_in, const int* in_sizes, int n_in,
                              void* d_out, int out_size, void* d_ws, size_t ws_size,
                              hipStream_t stream) {
    const float* h_in = (const float*)d_in[0];
    const float* ef   = (const float*)d_in[1];
    const int*   dst  = (const int*)d_in[2];
    const float* W    = (const float*)d_in[3];
    float* out = (float*)d_out;

    int N = in_sizes[0] / IN_SIZE;   // 50000
    int E = in_sizes[2];             // 1,600,000

    float* ws = (float*)d_ws;
    float* deg      = ws;
    float* sim_cnt  = ws + (size_t)N;
    float* esum     = ws + (size_t)2  * N;   // becomes agg
    float* hk       = ws + (size_t)34 * N;
    float* sim_sum  = ws + (size_t)66 * N;
    float* hk_norm  = ws + (size_t)98 * N;

    long ztot = (long)99 * N;
    k_zero<<<(int)((ztot + 255) / 256), 256, 0, stream>>>(ws, ztot);

    k_scatter1<<<(E + 255) / 256, 256, 0, stream>>>(ef, dst, deg, esum, E);
    k_hk<<<(N + 255) / 256, 256, 0, stream>>>(deg, esum, hk, hk_norm, N);
    k_scatter2<<<(E + 255) / 256, 256, 0, stream>>>(ef, dst, hk, hk_norm, sim_sum, sim_cnt, E);
    k_agg<<<(N + 255) / 256, 256, 0, stream>>>(sim_sum, sim_cnt, hk, esum, N);

    k_gemm_wmma<<<N / 16, 256, 0, stream>>>(h_in, esum /*agg*/, W, out, N);
}